// AttentionPooling_45535243272659
// MI455X (gfx1250) — compile-verified
//
#include <hip/hip_runtime.h>

#define N_NODES 262144
#define HID     512
#define NSEG    4096
#define NBLK2   1024   // N_NODES / 256

typedef float v2f __attribute__((ext_vector_type(2)));
typedef float v8f __attribute__((ext_vector_type(8)));

// Monotonic uint encoding of float (for exact, order-independent atomic max).
__device__ __forceinline__ unsigned fkey(float f) {
    unsigned b = __float_as_uint(f);
    return (b & 0x80000000u) ? ~b : (b | 0x80000000u);
}
__device__ __forceinline__ float funkey(unsigned k) {
    unsigned b = (k & 0x80000000u) ? (k & 0x7fffffffu) : ~k;
    return __uint_as_float(b);
}

__global__ void init_kernel(unsigned* Mkey) {
    if (threadIdx.x == 0) *Mkey = 0u;   // decodes below any real logit key
}

// Pass 1: logits = x @ W + b via V_WMMA_F32_16X16X4_F32 (exact fp32 GEMV).
// A-matrix (16x4) = W chunk replicated into all 16 rows; B-matrix (4x16) = x tile^T.
// Every lane of c[0] then holds the logit of row (tile*16 + (lane&15)).
__global__ void logits_kernel(const float* __restrict__ x,
                              const float* __restrict__ W,
                              const float* __restrict__ bias,
                              float* __restrict__ logits,
                              unsigned* __restrict__ Mkey) {
    __shared__ float red[128];
    const int lane  = threadIdx.x & 31;
    const int wave  = threadIdx.x >> 5;
    const int tile  = blockIdx.x * 8 + wave;        // 2048 blocks * 8 waves = 16384 tiles
    const int n     = lane & 15;                    // column (row of x) within tile
    const int khalf = lane >> 4;                    // VGPR0 holds K=0|2, VGPR1 K=1|3
    const float* xrow = x + (size_t)(tile * 16 + n) * HID + 2 * khalf;
    const float* wptr = W + 2 * khalf;

    v8f c = {0.f, 0.f, 0.f, 0.f, 0.f, 0.f, 0.f, 0.f};
#pragma unroll 4
    for (int k0 = 0; k0 < HID; k0 += 4) {
        v2f a  = *(const v2f*)(wptr + k0);          // W[k0+2h], W[k0+2h+1] (uniform per khalf)
        v2f bb = *(const v2f*)(xrow + k0);          // x[row][k0+2h], x[row][k0+2h+1]
        c = __builtin_amdgcn_wmma_f32_16x16x4_f32(false, a, false, bb,
                                                  (short)0, c, false, false);
    }
    float lg = c[0] + bias[0];
    if (lane < 16) {
        logits[tile * 16 + lane] = lg;
        red[wave * 16 + lane] = lg;
    }
    __syncthreads();
    for (int off = 64; off > 0; off >>= 1) {
        if (threadIdx.x < off)
            red[threadIdx.x] = fmaxf(red[threadIdx.x], red[threadIdx.x + off]);
        __syncthreads();
    }
    if (threadIdx.x == 0) atomicMax(Mkey, fkey(red[0]));   // exact max -> deterministic
}

// Pass 2a: per-block partial sums of exp(logit - M), fixed order.
__global__ void expsum_kernel(const float* __restrict__ logits,
                              const unsigned* __restrict__ Mkey,
                              float* __restrict__ partials) {
    __shared__ float red[256];
    const int i = blockIdx.x * 256 + threadIdx.x;
    const float M = funkey(*Mkey);
    red[threadIdx.x] = __expf(logits[i] - M);
    __syncthreads();
    for (int off = 128; off > 0; off >>= 1) {
        if (threadIdx.x < off) red[threadIdx.x] += red[threadIdx.x + off];
        __syncthreads();
    }
    if (threadIdx.x == 0) partials[blockIdx.x] = red[0];
}

// Pass 2b: single-block deterministic reduce of partials -> S; also decode M.
__global__ void finalize_kernel(const float* __restrict__ partials,
                                const unsigned* __restrict__ Mkey,
                                float* __restrict__ MS) {
    __shared__ float red[256];
    float s = 0.f;
    for (int j = threadIdx.x; j < NBLK2; j += 256) s += partials[j];
    red[threadIdx.x] = s;
    __syncthreads();
    for (int off = 128; off > 0; off >>= 1) {
        if (threadIdx.x < off) red[threadIdx.x] += red[threadIdx.x + off];
        __syncthreads();
    }
    if (threadIdx.x == 0) { MS[0] = funkey(*Mkey); MS[1] = red[0]; }
}

// Pass 3: segment start offsets (batch is sorted, every id appears).
__global__ void bounds_kernel(const int* __restrict__ batch,
                              int* __restrict__ seg_start) {
    const int i = blockIdx.x * 256 + threadIdx.x;
    const int b = batch[i];
    if (i == 0 || batch[i - 1] != b) seg_start[b] = i;
}

// Pass 4: one block per segment. s_i = exp(l_i - M)/S, then per-segment softmax
// of s, then out[b] = sum a_i * x[i]. All reductions in fixed order.
__global__ void pool_kernel(const float* __restrict__ x,
                            const float* __restrict__ logits,
                            const int* __restrict__ seg_start,
                            const float* __restrict__ MS,
                            float* __restrict__ out) {
    __shared__ float red[256];
    const int b = blockIdx.x;
    const int t = threadIdx.x;
    const int start = seg_start[b];
    const int end   = (b == NSEG - 1) ? N_NODES : seg_start[b + 1];
    const float M    = MS[0];
    const float invS = 1.0f / MS[1];

    // per-segment max of logits (monotonic with s) -> smax in s-domain
    float lmax = -3.402823466e38f;
    for (int i = start + t; i < end; i += 256) lmax = fmaxf(lmax, logits[i]);
    red[t] = lmax;
    __syncthreads();
    for (int off = 128; off > 0; off >>= 1) {
        if (t < off) red[t] = fmaxf(red[t], red[t + off]);
        __syncthreads();
    }
    const float smax = __expf(red[0] - M) * invS;
    __syncthreads();

    // denominator: sum exp(s_i - smax), fixed-order strided + tree
    float ds = 0.f;
    for (int i = start + t; i < end; i += 256) {
        float s = __expf(logits[i] - M) * invS;
        ds += __expf(s - smax);
    }
    red[t] = ds;
    __syncthreads();
    for (int off = 128; off > 0; off >>= 1) {
        if (t < off) red[t] += red[t + off];
        __syncthreads();
    }
    const float invD = 1.0f / red[0];

    // weighted sum: each thread owns 2 columns (256*2 = 512), float2 coalesced
    float acc0 = 0.f, acc1 = 0.f;
    const float* xcol = x + 2 * t;
    for (int i = start; i < end; ++i) {
        float s = __expf(logits[i] - M) * invS;
        float a = __expf(s - smax) * invD;
        v2f xv = *(const v2f*)(xcol + (size_t)i * HID);
        acc0 += xv.x * a;
        acc1 += xv.y * a;
    }
    out[(size_t)b * HID + 2 * t]     = acc0;
    out[(size_t)b * HID + 2 * t + 1] = acc1;
}

extern "C" void kernel_launch(void* const* d_in, const int* in_sizes, int n_in,
                              void* d_out, int out_size, void* d_ws, size_t ws_size,
                              hipStream_t stream) {
    const float* x     = (const float*)d_in[0];   // [N, 512]
    const int*   batch = (const int*)d_in[1];     // [N] sorted segment ids
    const float* W     = (const float*)d_in[2];   // [512]
    const float* bias  = (const float*)d_in[3];   // [1]
    float*       out   = (float*)d_out;           // [4096, 512]

    // Workspace layout (~1.07 MB total)
    float*    logits    = (float*)d_ws;               // N floats
    float*    partials  = logits + N_NODES;           // NBLK2 floats
    int*      seg_start = (int*)(partials + NBLK2);   // NSEG ints
    unsigned* Mkey      = (unsigned*)(seg_start + NSEG);
    float*    MS        = (float*)(Mkey + 2);         // [M, S]

    init_kernel<<<1, 1, 0, stream>>>(Mkey);
    logits_kernel<<<N_NODES / 128, 256, 0, stream>>>(x, W, bias, logits, Mkey);
    expsum_kernel<<<NBLK2, 256, 0, stream>>>(logits, Mkey, partials);
    finalize_kernel<<<1, 256, 0, stream>>>(partials, Mkey, MS);
    bounds_kernel<<<N_NODES / 256, 256, 0, stream>>>(batch, seg_start);
    pool_kernel<<<NSEG, 256, 0, stream>>>(x, logits, seg_start, MS, out);
}